// Fold_31980326486781
// MI455X (gfx1250) — compile-verified
//
#include <hip/hip_runtime.h>
#include <cstdint>

// Fold (col2im): cols (16,64,128,128,3,3) f32 -> out (16,64,128,128) f32
// stride=1, pad=1, dilation=1.
// out[b,c,y,x] = sum_{i,j} cols[b,c, y+1-i, x+1-j, i, j]  (in-range terms only)
//
// Memory-bound: 604 MB in + 67 MB out -> ~29 us floor @ 23.3 TB/s.
// Per block: async-DMA 10 cols rows (8 + 2 halo) into LDS via CDNA5
// GLOBAL_LOAD_ASYNC_TO_LDS_B128 (ASYNCcnt), gather stride-9 from LDS
// (bank-conflict-free, gcd(9,64)=1), store coalesced b32.
// The async op applies its immediate offset to BOTH the global and LDS
// address, so the 8 contiguous interior rows need just one address pair
// per thread + 9 immediate offsets.

#define LH 128
#define LW 128
#define ROW_FLOATS (LW * 9)          // 1152 floats per cols row
#define ROW_BYTES  (ROW_FLOATS * 4)  // 4608 bytes (16B multiple)
#define TILE_ROWS  8                 // output rows per block
#define LDS_ROWS   (TILE_ROWS + 2)   // + top/bottom halo rows
#define THREADS    256

#define ASYNC_CP(la, ga)        \
    asm volatile("global_load_async_to_lds_b128 %0, %1, off" :: "v"(la), "v"(ga) : "memory")
#define ASYNC_CP_OFF(la, ga, O) \
    asm volatile("global_load_async_to_lds_b128 %0, %1, off offset:" #O :: "v"(la), "v"(ga) : "memory")

__global__ __launch_bounds__(THREADS)
void fold_gather_kernel(const float* __restrict__ cols, float* __restrict__ out) {
    __shared__ __align__(16) float smem[LDS_ROWS * ROW_FLOATS]; // 46,080 B

    const int tid   = threadIdx.x;
    const int plane = blockIdx.x >> 4;      // b*64 + c  (1024 planes)
    const int tile  = blockIdx.x & 15;      // 16 row-tiles per plane
    const int y0    = tile * TILE_ROWS;

    const char* gplane = (const char*)cols + (size_t)plane * (size_t)(LH * ROW_BYTES);
    const uint32_t lds_base = (uint32_t)(uintptr_t)(void*)&smem[0]; // LDS byte offset
    const uint32_t toff = (uint32_t)(tid * 16);

    // ---- Interior rows r=1..8 (lh = y0..y0+7, always valid) ----
    // 8 rows * 4608 B = 36,864 B contiguous in both global and LDS:
    // 2304 16B-chunks = exactly 9 per thread, via immediate offsets k*4096.
    {
        uint64_t ga = (uint64_t)(uintptr_t)(gplane + (size_t)y0 * ROW_BYTES) + toff;
        uint32_t la = lds_base + (uint32_t)ROW_BYTES + toff;
        asm volatile(
            "global_load_async_to_lds_b128 %0, %1, off\n\t"
            "global_load_async_to_lds_b128 %0, %1, off offset:4096\n\t"
            "global_load_async_to_lds_b128 %0, %1, off offset:8192\n\t"
            "global_load_async_to_lds_b128 %0, %1, off offset:12288\n\t"
            "global_load_async_to_lds_b128 %0, %1, off offset:16384\n\t"
            "global_load_async_to_lds_b128 %0, %1, off offset:20480\n\t"
            "global_load_async_to_lds_b128 %0, %1, off offset:24576\n\t"
            "global_load_async_to_lds_b128 %0, %1, off offset:28672\n\t"
            "global_load_async_to_lds_b128 %0, %1, off offset:32768"
            :: "v"(la), "v"(ga) : "memory");
    }

    // ---- Top halo r=0 (lh = y0-1): block-uniform validity ----
    {
        const uint32_t la = lds_base + toff;
        if (tile > 0) {
            uint64_t ga = (uint64_t)(uintptr_t)(gplane + (size_t)(y0 - 1) * ROW_BYTES) + toff;
            ASYNC_CP(la, ga);                          // chunks 0..255
            if (tid < 32) ASYNC_CP_OFF(la, ga, 4096);  // chunks 256..287 (wave 0)
        } else {
            const float4 z = make_float4(0.f, 0.f, 0.f, 0.f);
            float4* dst = (float4*)&smem[0];
            dst[tid] = z;
            if (tid < 32) dst[tid + 256] = z;
        }
    }

    // ---- Bottom halo r=9 (lh = y0+8): block-uniform validity ----
    {
        const uint32_t la = lds_base + (uint32_t)(9 * ROW_BYTES) + toff;
        if (tile < 15) {
            uint64_t ga = (uint64_t)(uintptr_t)(gplane + (size_t)(y0 + 8) * ROW_BYTES) + toff;
            ASYNC_CP(la, ga);
            if (tid < 32) ASYNC_CP_OFF(la, ga, 4096);
        } else {
            const float4 z = make_float4(0.f, 0.f, 0.f, 0.f);
            float4* dst = (float4*)&smem[9 * ROW_FLOATS];
            dst[tid] = z;
            if (tid < 32) dst[tid + 256] = z;
        }
    }

    asm volatile("s_wait_asynccnt 0" ::: "memory"); // our async copies complete
    __syncthreads();                                 // + LDS zero-fills visible

    // ---- Gather: each thread produces 4 outputs (fixed x, 4 rows) ----
    const int x    = tid & (LW - 1);  // 0..127 (known bits)
    const int half = tid >> 7;        // 0 or 1
    float* oplane = out + (size_t)plane * (LH * LW) + (size_t)y0 * LW;

    #pragma unroll
    for (int k = 0; k < TILE_ROWS / 2; ++k) {
        const int ry = half + 2 * k;  // output row within tile, 0..7
        float acc = 0.f;
        #pragma unroll
        for (int i = 0; i < 3; ++i) {
            // source cols row lh = (y0+ry)+1-i -> LDS row ry+2-i (halo zeroed)
            const float* row = &smem[(ry + 2 - i) * ROW_FLOATS + i * 3];
            if (x + 1 < LW) acc += row[(x + 1) * 9 + 0]; // j=0, lw=x+1
            acc += row[x * 9 + 1];                        // j=1, lw=x (always ok)
            if (x >= 1)     acc += row[(x - 1) * 9 + 2]; // j=2, lw=x-1
        }
        oplane[ry * LW + x] = acc; // coalesced b32 store
    }
}

extern "C" void kernel_launch(void* const* d_in, const int* in_sizes, int n_in,
                              void* d_out, int out_size, void* d_ws, size_t ws_size,
                              hipStream_t stream) {
    const float* cols = (const float*)d_in[0];
    float* out = (float*)d_out;
    // 1024 planes * 16 row-tiles = 16384 blocks of 256 threads (8 waves, wave32)
    dim3 grid(16 * 64 * (LH / TILE_ROWS));
    fold_gather_kernel<<<grid, dim3(THREADS), 0, stream>>>(cols, out);
}